// LightGCN_10290741641399
// MI455X (gfx1250) — compile-verified
//
#include <hip/hip_runtime.h>
#include <hip/hip_bf16.h>

#define EMB_DIM 64

typedef __attribute__((ext_vector_type(2))) float v2f;
typedef __attribute__((ext_vector_type(8))) float v8f;

// ---------------------------------------------------------------------------
// init: out = emb, xcur = emb, xnext = 0   (float4 vectorized, grid-stride)
// ---------------------------------------------------------------------------
__global__ void lgcn_init(const float* __restrict__ emb,
                          float* __restrict__ xcur,
                          float* __restrict__ xnext,
                          float* __restrict__ out,
                          long n4) {
  long i = (long)blockIdx.x * blockDim.x + threadIdx.x;
  long stride = (long)gridDim.x * blockDim.x;
  const float4* e4 = (const float4*)emb;
  float4* c4 = (float4*)xcur;
  float4* x4 = (float4*)xnext;
  float4* o4 = (float4*)out;
  const float4 z = make_float4(0.f, 0.f, 0.f, 0.f);
  for (; i < n4; i += stride) {
    float4 v = e4[i];
    c4[i] = v;
    o4[i] = v;
    x4[i] = z;
  }
}

// ---------------------------------------------------------------------------
// scatter: xnext[dst] += xcur[src]  (one wave per edge, lane covers 2 dims)
// Coalesced 256B row read (32 lanes x float2), global_atomic_add_f32 writes.
// global_prefetch_b8 on the next edge's source row hides random latency.
// ---------------------------------------------------------------------------
__global__ void lgcn_scatter(const float* __restrict__ xcur,
                             float* __restrict__ xnext,
                             const int* __restrict__ src,
                             const int* __restrict__ dst,
                             int nEdges, int nNodes) {
  const int lane   = threadIdx.x & 31;
  const int wave   = ((int)blockIdx.x * (int)blockDim.x + (int)threadIdx.x) >> 5;
  const int nWaves = ((int)gridDim.x * (int)blockDim.x) >> 5;
  for (int e = wave; e < nEdges; e += nWaves) {
    const int s = src[e];
    const int d = dst[e];
    // Prefetch the next edge's source row into cache (gfx1250 global_prefetch_b8)
    const int en = e + nWaves;
    if (en < nEdges) {
      const int sn = src[en];
      if ((unsigned)sn < (unsigned)nNodes)
        __builtin_prefetch(xcur + (long)sn * EMB_DIM + lane * 2, 0, 1);
    }
    if ((unsigned)s >= (unsigned)nNodes || (unsigned)d >= (unsigned)nNodes) continue;
    const float2 v = ((const float2*)(xcur + (long)s * EMB_DIM))[lane];
    float* o = xnext + (long)d * EMB_DIM + lane * 2;
    atomicAdd(o + 0, v.x);   // global_atomic_add_f32
    atomicAdd(o + 1, v.y);
  }
}

// ---------------------------------------------------------------------------
// accum: out += xnext; xcur = xnext; xnext = 0   (fused, float4)
// ---------------------------------------------------------------------------
__global__ void lgcn_accum(float* __restrict__ out,
                           float* __restrict__ xcur,
                           float* __restrict__ xnext,
                           long n4) {
  long i = (long)blockIdx.x * blockDim.x + threadIdx.x;
  long stride = (long)gridDim.x * blockDim.x;
  float4* o4 = (float4*)out;
  float4* c4 = (float4*)xcur;
  float4* x4 = (float4*)xnext;
  const float4 z = make_float4(0.f, 0.f, 0.f, 0.f);
  for (; i < n4; i += stride) {
    float4 nv = x4[i];
    float4 ov = o4[i];
    ov.x += nv.x; ov.y += nv.y; ov.z += nv.z; ov.w += nv.w;
    o4[i] = ov;
    c4[i] = nv;
    x4[i] = z;
  }
}

// ---------------------------------------------------------------------------
// scale: out = out * 0.25 computed as out x (0.25*I64) via V_WMMA_F32_16X16X4_F32.
// Bit-exact (0.25 = 2^-2; identity kills all cross terms). One wave per 16x16
// tile; 4 chained WMMAs cover the K-slice [n0, n0+16). Tiles disjoint -> safe
// in place. Wave-uniform bounds keep EXEC all-ones (WMMA requirement).
// ---------------------------------------------------------------------------
__global__ void lgcn_scale_wmma(float* __restrict__ out, int nRows) {
  const int lane   = threadIdx.x & 31;
  const int wave   = ((int)blockIdx.x * (int)blockDim.x + (int)threadIdx.x) >> 5;
  const int nWaves = ((int)gridDim.x * (int)blockDim.x) >> 5;
  const int nColTiles = EMB_DIM / 16;            // 4
  const int nTiles    = (nRows / 16) * nColTiles;
  const int half = lane >> 4;                    // 0: lanes 0-15, 1: lanes 16-31
  const int ln   = lane & 15;

  for (int t = wave; t < nTiles; t += nWaves) {
    const int m0 = (t / nColTiles) * 16;         // row tile origin
    const int n0 = (t % nColTiles) * 16;         // col tile origin
    v8f c = {};
    #pragma unroll
    for (int kk = 0; kk < 16; kk += 4) {
      const int k0 = n0 + kk;
      // A (16x4 f32): lane holds rows m0+ln, cols k0 + 2*half + {0,1}
      const float* ap = out + (long)(m0 + ln) * EMB_DIM + k0 + 2 * half;
      v2f a;
      a.x = ap[0];
      a.y = ap[1];
      // B (4x16): 0.25 where global k == global n, i.e. row r hits col (kk + r)
      const int r0 = 2 * half;                   // v0 row; v1 row = r0+1
      v2f b;
      b.x = (ln == kk + r0)     ? 0.25f : 0.0f;
      b.y = (ln == kk + r0 + 1) ? 0.25f : 0.0f;
      c = __builtin_amdgcn_wmma_f32_16x16x4_f32(
              /*neg_a=*/false, a, /*neg_b=*/false, b,
              /*c_mod=*/(short)0, c, /*reuse_a=*/false, /*reuse_b=*/false);
    }
    // D layout: VGPR i -> row m0 + i + 8*half, col n0 + ln
    #pragma unroll
    for (int i = 0; i < 8; ++i) {
      out[(long)(m0 + i + 8 * half) * EMB_DIM + (n0 + ln)] = c[i];
    }
  }
}

// ---------------------------------------------------------------------------
// Launch: inputs are [embedding f32 (N*64), edge_index int (2*E), num_users]
// d_out = 200000*64 f32 (user || item concatenated == final flat).
// ---------------------------------------------------------------------------
extern "C" void kernel_launch(void* const* d_in, const int* in_sizes, int n_in,
                              void* d_out, int out_size, void* d_ws, size_t ws_size,
                              hipStream_t stream) {
  const float* emb  = (const float*)d_in[0];
  const int*   ei   = (const int*)d_in[1];
  const int    N    = in_sizes[0] / EMB_DIM;     // 200000
  const int    E    = in_sizes[1] / 2;           // 1250000
  const int*   src  = ei;
  const int*   dst  = ei + E;
  float*       out  = (float*)d_out;

  float* xcur  = (float*)d_ws;
  float* xnext = xcur + (long)N * EMB_DIM;

  const long n4 = (long)N * EMB_DIM / 4;

  lgcn_init<<<4096, 256, 0, stream>>>(emb, xcur, xnext, out, n4);

  for (int layer = 0; layer < 3; ++layer) {
    lgcn_scatter<<<2048, 256, 0, stream>>>(xcur, xnext, src, dst, E, N);
    lgcn_accum<<<4096, 256, 0, stream>>>(out, xcur, xnext, n4);
  }

  lgcn_scale_wmma<<<2048, 256, 0, stream>>>(out, N);
}